// Encoder_77369540870268
// MI455X (gfx1250) — compile-verified
//
#include <hip/hip_runtime.h>
#include <math.h>

typedef __attribute__((ext_vector_type(2))) float v2f;
typedef __attribute__((ext_vector_type(8))) float v8f;

#define B_      8
#define P0_     20000
#define P1_     5000
#define P2_     1250
#define M_      10
#define W_      9
#define C0_     3
#define C1_     64
#define C2_     128
#define LATENT_ 256
#define KFIN    (P2_*C2_)        /* 160000 */
#define KSPLIT  125
#define KCHUNK  (KFIN / KSPLIT)  /* 1280 */
#define K1P     28               /* 27 (=W*C0) padded to mult of 4 */
#define C1P     66               /* 64 padded to kill LDS bank conflicts */

__device__ __forceinline__ float eluf(float z) {
    return z > 0.0f ? z : (__expf(z) - 1.0f);
}

__device__ __forceinline__ v8f wmma4(v2f a, v2f b, v8f c) {
    // D = A(16x4 f32) * B(4x16 f32) + C(16x16 f32)
    return __builtin_amdgcn_wmma_f32_16x16x4_f32(false, a, false, b, (short)0, c,
                                                 false, false);
}

// ---------------------------------------------------------------------------
// Conv layer 1: rows = b*P0+p (160000), K = 27 (W*C0, padded to 28), N = 64.
// t0[row,(w,i)] = sum_m ww0[p,m,w] * x[b, nb0[p,m], i]   (nb==P0 -> zero)
// h1[row,o]     = elu( sum_k t0[row,k] * W0r[k,o] + bias0[o] )
// B staged N-major: fragment (B[k][o],B[k+1][o]) is one contiguous b64 load.
// Row stride 28 floats: 28*r mod 64 distinct over 16 rows -> conflict-free.
// ---------------------------------------------------------------------------
__global__ __launch_bounds__(128)
void conv1_kernel(const float* __restrict__ x, const int* __restrict__ nb0,
                  const float* __restrict__ w0, const float* __restrict__ bias0,
                  const float* __restrict__ ww0, float* __restrict__ h1)
{
    __shared__ float sT[64][K1P];      // A tile: 64 rows x 28
    __shared__ float sB[C1_][K1P];     // B tile, N-major: 64 outs x 28 K

    const int t = threadIdx.x;
    const int rowBase = blockIdx.x * 64;

    // Stage B: sB[o][w*3+i] = weights0[w][o*3+i]; K=27 padded with zero.
    for (int idx = t; idx < C1_ * K1P; idx += 128) {
        int o = idx / K1P, k = idx - o * K1P;
        float v = 0.0f;
        if (k < W_ * C0_) {
            int w = k / C0_, i = k - w * C0_;
            v = w0[w * (C1_ * C0_) + o * C0_ + i];
        }
        sB[o][k] = v;
    }

    // Stage A: gather + contract over m.
    for (int idx = t; idx < 64 * K1P; idx += 128) {
        int r = idx / K1P, k = idx - r * K1P;
        float acc = 0.0f;
        if (k < W_ * C0_) {
            int w = k / C0_, i = k - w * C0_;
            int gr = rowBase + r;
            int b = gr / P0_, p = gr - b * P0_;
            const int*   nbp = nb0 + p * M_;
            const float* wwp = ww0 + (size_t)p * (M_ * W_) + w;
            #pragma unroll
            for (int m = 0; m < M_; ++m) {
                int nb = nbp[m];
                if (nb != P0_)
                    acc += wwp[m * W_] * x[((size_t)b * P0_ + nb) * C0_ + i];
            }
        }
        sT[r][k] = acc;
    }
    __syncthreads();

    const int wave = t >> 5, lane = t & 31;
    const int lr = lane & 15;
    const int kh = (lane >> 4) * 2;
    const int rs = (lane >> 4) * 8;

    v8f acc[4] = {};
    for (int k = 0; k < K1P; k += 4) {
        v2f a;
        a.x = sT[wave * 16 + lr][k + kh];
        a.y = sT[wave * 16 + lr][k + kh + 1];
        #pragma unroll
        for (int nt = 0; nt < 4; ++nt) {
            v2f b;
            b.x = sB[nt * 16 + lr][k + kh];
            b.y = sB[nt * 16 + lr][k + kh + 1];
            acc[nt] = wmma4(a, b, acc[nt]);
        }
    }
    #pragma unroll
    for (int nt = 0; nt < 4; ++nt) {
        int o = nt * 16 + lr;
        float bia = bias0[o];
        #pragma unroll
        for (int j = 0; j < 8; ++j) {
            int gr = rowBase + wave * 16 + j + rs;
            h1[(size_t)gr * C1_ + o] = eluf(acc[nt][j] + bia);
        }
    }
}

// ---------------------------------------------------------------------------
// Pool: hout[b,q,c] = elu( sum_m p[m]*hin[b,nb[q,m],c] / (sum_m p[m] + 1e-8) )
// with p[m] = |pw[q,m]| masked by nb != Pin. block = C threads, grid=(Pout,B)
// ---------------------------------------------------------------------------
__global__ void pool_kernel(const float* __restrict__ hin, const int* __restrict__ nb,
                            const float* __restrict__ pw, float* __restrict__ hout,
                            int Pin, int Pout, int C)
{
    int c = threadIdx.x;
    int q = blockIdx.x;
    int b = blockIdx.y;
    const int*   nbp = nb + q * M_;
    const float* pwp = pw + q * M_;
    float den = 1e-8f, num = 0.0f;
    #pragma unroll
    for (int m = 0; m < M_; ++m) {
        int n = nbp[m];
        if (n != Pin) {
            float p = fabsf(pwp[m]);
            den += p;
            num += p * hin[((size_t)b * Pin + n) * C + c];
        }
    }
    hout[((size_t)b * Pout + q) * C + c] = eluf(num / den);
}

// ---------------------------------------------------------------------------
// Conv layer 2: rows = b*P1+q (40000), K = 576 chunked by w (9 x 64), N = 128.
// per chunk w:  t[row,i] = sum_m ww2[q,m,w]*h2[b,nb2[q,m],i]
//               B (N-major): sB[o][i] = weights2[w][o*64+i]  (contiguous copy)
// Rows padded to 66 floats -> per-lane banks 2*row mod 64, conflict-free b64.
// ---------------------------------------------------------------------------
__global__ __launch_bounds__(128)
void conv3_kernel(const float* __restrict__ h2, const int* __restrict__ nb2,
                  const float* __restrict__ w2, const float* __restrict__ bias2,
                  const float* __restrict__ ww2, float* __restrict__ h3)
{
    __shared__ float sT[64][C1P];      // ~16.5 KB
    __shared__ float sB[C2_][C1P];     // ~33   KB, N-major

    const int t = threadIdx.x;
    const int rowBase = blockIdx.x * 64;
    const int wave = t >> 5, lane = t & 31;
    const int lr = lane & 15;
    const int kh = (lane >> 4) * 2;
    const int rs = (lane >> 4) * 8;

    v8f acc[8] = {};
    for (int w = 0; w < W_; ++w) {
        // B chunk: straight copy of weights2 row w (o-major, i contiguous).
        for (int idx = t; idx < C2_ * C1_; idx += 128) {
            int o = idx >> 6, i = idx & 63;
            sB[o][i] = w2[(size_t)w * (C2_ * C1_) + idx];
        }
        // A chunk: gather + contract over m.
        for (int idx = t; idx < 64 * C1_; idx += 128) {
            int r = idx >> 6, i = idx & 63;
            int gr = rowBase + r;
            int b = gr / P1_, q = gr - b * P1_;
            const int*   nbp = nb2 + q * M_;
            const float* wwp = ww2 + (size_t)q * (M_ * W_) + w;
            float s = 0.0f;
            #pragma unroll
            for (int m = 0; m < M_; ++m) {
                int nb = nbp[m];
                if (nb != P1_)
                    s += wwp[m * W_] * h2[((size_t)b * P1_ + nb) * C1_ + i];
            }
            sT[r][i] = s;
        }
        __syncthreads();
        for (int k = 0; k < C1_; k += 4) {
            v2f a;
            a.x = sT[wave * 16 + lr][k + kh];
            a.y = sT[wave * 16 + lr][k + kh + 1];
            #pragma unroll
            for (int nt = 0; nt < 8; ++nt) {
                v2f b;
                b.x = sB[nt * 16 + lr][k + kh];
                b.y = sB[nt * 16 + lr][k + kh + 1];
                acc[nt] = wmma4(a, b, acc[nt]);
            }
        }
        __syncthreads();
    }
    #pragma unroll
    for (int nt = 0; nt < 8; ++nt) {
        int o = nt * 16 + lr;
        float bia = bias2[o];
        #pragma unroll
        for (int j = 0; j < 8; ++j) {
            int gr = rowBase + wave * 16 + j + rs;
            h3[(size_t)gr * C2_ + o] = eluf(acc[nt][j] + bia);
        }
    }
}

// ---------------------------------------------------------------------------
// Final dense: out[b,j] = f[b,:] . Wrow(j) + bias(j).  M=8 (in 16-row WMMA),
// N = 512 (8 j-groups of 64), split-K over 125 chunks of 1280. Deterministic
// partials -> reduce kernel. Weights streamed once (327 MB, BW-bound).
// ---------------------------------------------------------------------------
__global__ __launch_bounds__(256)
void final_gemm_kernel(const float* __restrict__ f, const float* __restrict__ Wm,
                       const float* __restrict__ Wsg, float* __restrict__ partial)
{
    const int t = threadIdx.x;
    const int wave = t >> 5, lane = t & 31;
    const int task = blockIdx.x * 8 + wave;      // 0..999
    const int jgroup = task & 7;                 // 0..7  -> 64 outputs
    const int chunk  = task >> 3;                // 0..124
    const int lr = lane & 15;
    const int kh = (lane >> 4) * 2;
    const int j0 = jgroup * 64;

    const float* wp[4];
    #pragma unroll
    for (int jt = 0; jt < 4; ++jt) {
        int j = j0 + jt * 16 + lr;
        wp[jt] = (j < LATENT_) ? (Wm + (size_t)j * KFIN)
                               : (Wsg + (size_t)(j - LATENT_) * KFIN);
    }
    const int row = lr;  // batch index for A (rows >= 8 are zero-padding)
    const float* fp = f + (size_t)(row < B_ ? row : 0) * KFIN;

    v8f acc[4] = {};
    const int k0 = chunk * KCHUNK;
    for (int kb = k0; kb < k0 + KCHUNK; kb += 64) {
        #pragma unroll
        for (int jt = 0; jt < 4; ++jt)
            __builtin_prefetch((const void*)(wp[jt] + kb + 256), 0, 1);
        for (int k = kb; k < kb + 64; k += 4) {
            v2f a;
            if (row < B_) {
                a.x = fp[k + kh];
                a.y = fp[k + kh + 1];
            } else {
                a.x = 0.0f; a.y = 0.0f;
            }
            #pragma unroll
            for (int jt = 0; jt < 4; ++jt) {
                v2f b;
                b.x = wp[jt][k + kh];
                b.y = wp[jt][k + kh + 1];
                acc[jt] = wmma4(a, b, acc[jt]);
            }
        }
    }
    // D layout: lanes 0..15 hold rows 0..7 in regs 0..7 -> exactly batches 0..7
    if (lane < 16) {
        #pragma unroll
        for (int jt = 0; jt < 4; ++jt) {
            int j = j0 + jt * 16 + lr;
            #pragma unroll
            for (int b = 0; b < 8; ++b)
                partial[((size_t)chunk * 512 + j) * 8 + b] = acc[jt][b];
        }
    }
}

__global__ __launch_bounds__(256)
void reduce_kernel(const float* __restrict__ partial, const float* __restrict__ bm,
                   const float* __restrict__ bs, float* __restrict__ out)
{
    int tid = blockIdx.x * 256 + threadIdx.x;   // 0..4095
    int b = tid >> 9;
    int j = tid & 511;
    float s = (j < LATENT_) ? bm[j] : bs[j - LATENT_];
    for (int ch = 0; ch < KSPLIT; ++ch)
        s += partial[((size_t)ch * 512 + j) * 8 + b];
    out[b * 512 + j] = s;
}

// ---------------------------------------------------------------------------
extern "C" void kernel_launch(void* const* d_in, const int* in_sizes, int n_in,
                              void* d_out, int out_size, void* d_ws, size_t ws_size,
                              hipStream_t stream)
{
    (void)in_sizes; (void)n_in; (void)out_size; (void)ws_size;
    const float* x   = (const float*)d_in[0];
    const int*   nb0 = (const int*)  d_in[1];
    const int*   nb1 = (const int*)  d_in[2];
    const int*   nb2 = (const int*)  d_in[3];
    const int*   nb3 = (const int*)  d_in[4];
    const float* w0  = (const float*)d_in[5];
    const float* b0  = (const float*)d_in[6];
    const float* ww0 = (const float*)d_in[7];
    const float* pn1 = (const float*)d_in[8];
    const float* w2  = (const float*)d_in[9];
    const float* b2  = (const float*)d_in[10];
    const float* ww2 = (const float*)d_in[11];
    const float* pn3 = (const float*)d_in[12];
    const float* Wm  = (const float*)d_in[13];
    const float* bm  = (const float*)d_in[14];
    const float* Wsg = (const float*)d_in[15];
    const float* bs  = (const float*)d_in[16];

    float* ws = (float*)d_ws;
    size_t o = 0;
    float* h1   = ws + o; o += (size_t)B_ * P0_ * C1_;   // 10.24 M
    float* h2   = ws + o; o += (size_t)B_ * P1_ * C1_;   //  2.56 M
    float* h3   = ws + o; o += (size_t)B_ * P1_ * C2_;   //  5.12 M
    float* fbuf = ws + o; o += (size_t)B_ * KFIN;        //  1.28 M
    float* part = ws + o;                                 //  0.512 M

    conv1_kernel<<<dim3((B_ * P0_) / 64), dim3(128), 0, stream>>>(
        x, nb0, w0, b0, ww0, h1);

    pool_kernel<<<dim3(P1_, B_), dim3(C1_), 0, stream>>>(
        h1, nb1, pn1, h2, P0_, P1_, C1_);

    conv3_kernel<<<dim3((B_ * P1_) / 64), dim3(128), 0, stream>>>(
        h2, nb2, w2, b2, ww2, h3);

    pool_kernel<<<dim3(P2_, B_), dim3(C2_), 0, stream>>>(
        h3, nb3, pn3, fbuf, P1_, P2_, C2_);

    final_gemm_kernel<<<dim3(KSPLIT), dim3(256), 0, stream>>>(
        fbuf, Wm, Wsg, part);

    reduce_kernel<<<dim3(16), dim3(256), 0, stream>>>(
        part, bm, bs, (float*)d_out);
}